// ColumnParallelLinearWithTopping_21036749816229
// MI455X (gfx1250) — compile-verified
//
#include <hip/hip_runtime.h>
#include <hip/hip_bf16.h>
#include <stdint.h>

// ---------------- problem constants ----------------
#define T_TOK   8192
#define DIM     4096            // D_IN == D_OUT
#define N_ADAPT 8
#define RANK    16

// ---------------- tiling ----------------
#define M_TILE   128
#define N_TILE   128
#define K_CHUNK  16
#define MAX_TILES 72            // sum ceil(cnt_a/128) <= T/128 + N_ADAPT = 64+8

#define XS_S 20                 // xs row stride: mult of 4 (float4 stores, v2f reads),
                                // m*20 mod 64 -> 16 distinct banks for 16 lanes
#define BN_S 258                // bs k-pair row stride: 128 cols * 2 + pad, even (8B align)

// ---------------- workspace layout (bytes) ----------------
#define WS_COUNTS  0                       // int[8]
#define WS_OFFSETS 32                      // int[8]
#define WS_CURSOR  64                      // int[8]
#define WS_TILES   128                     // TileDesc[MAX_TILES]
#define WS_SORTED  4096                    // int[T_TOK]
#define WS_R       (4096 + T_TOK * 4)      // float[T_TOK*RANK]  (sorted order)

typedef __attribute__((ext_vector_type(2))) float v2f;
typedef __attribute__((ext_vector_type(8))) float v8f;

struct TileDesc { int adapter; int row_start; int row_cnt; int pad; };

// ============================================================
// 1) zero counters
// ============================================================
__global__ void k_zero(char* ws) {
    int* p = (int*)(ws + WS_COUNTS);
    if (threadIdx.x < 24) p[threadIdx.x] = 0;
}

// ============================================================
// 2) histogram of adapter ids
// ============================================================
__global__ void k_count(const int* __restrict__ idx, char* ws) {
    int t = blockIdx.x * blockDim.x + threadIdx.x;
    if (t < T_TOK) atomicAdd((int*)(ws + WS_COUNTS) + idx[t], 1);
}

// ============================================================
// 3) prefix scan (8 bins) + adapter-uniform tile descriptors
// ============================================================
__global__ void k_scan(char* ws) {
    int* counts = (int*)(ws + WS_COUNTS);
    int* offs   = (int*)(ws + WS_OFFSETS);
    int* cur    = (int*)(ws + WS_CURSOR);
    TileDesc* tiles = (TileDesc*)(ws + WS_TILES);
    int o = 0, t = 0;
    for (int a = 0; a < N_ADAPT; ++a) {
        offs[a] = o; cur[a] = o;
        int c = counts[a];
        for (int s = 0; s < c; s += M_TILE) {
            tiles[t].adapter   = a;
            tiles[t].row_start = o + s;
            tiles[t].row_cnt   = min(M_TILE, c - s);
            ++t;
        }
        o += c;
    }
    for (; t < MAX_TILES; ++t) tiles[t].row_cnt = 0;
}

// ============================================================
// 4) scatter token ids into adapter-sorted order
// ============================================================
__global__ void k_scatter(const int* __restrict__ idx, char* ws) {
    int t = blockIdx.x * blockDim.x + threadIdx.x;
    if (t >= T_TOK) return;
    int a = idx[t];
    int pos = atomicAdd((int*)(ws + WS_CURSOR) + a, 1);
    ((int*)(ws + WS_SORTED))[pos] = t;
}

// ============================================================
// 5) r[i,16] = x[tok(i)] @ A[adapter]  (WMMA, N = RANK)
//    branchless: row index clamped; invalid rows only corrupt
//    their own (never-stored) output rows.
// ============================================================
__global__ __launch_bounds__(256) void k_rank(const float* __restrict__ x,
                                              const float* __restrict__ Abuf,
                                              char* ws) {
    TileDesc td = ((const TileDesc*)(ws + WS_TILES))[blockIdx.x];
    if (td.row_cnt == 0) return;
    const int* sorted = (const int*)(ws + WS_SORTED);
    float* r = (float*)(ws + WS_R);

    const int lane  = threadIdx.x & 31;
    const int wv    = threadIdx.x >> 5;                 // row-tile 0..7
    const int mrow  = wv * 16 + (lane & 15);
    const int srow  = td.row_start + min(mrow, td.row_cnt - 1);
    const long tok  = (long)sorted[srow];
    const int khalf = (lane >> 4) << 1;                 // 0 or 2
    const float* xr = x + tok * DIM + khalf;
    const float* ab = Abuf + (long)td.adapter * DIM * RANK
                           + (long)khalf * RANK + (lane & 15);

    v8f acc = {0.f,0.f,0.f,0.f,0.f,0.f,0.f,0.f};
    #pragma unroll 4
    for (int k = 0; k < DIM; k += 4) {
        v2f a = *(const v2f*)(xr + k);                  // 8B aligned
        v2f b;
        b.x = ab[(long)k * RANK];
        b.y = ab[(long)k * RANK + RANK];
        acc = __builtin_amdgcn_wmma_f32_16x16x4_f32(false, a, false, b,
                                                    (short)0, acc, false, false);
    }
    const int j  = lane & 15;
    const int mh = (lane >> 4) << 3;
    #pragma unroll
    for (int v = 0; v < 8; ++v) {
        int m2 = wv * 16 + v + mh;
        if (m2 < td.row_cnt)
            r[(long)(td.row_start + m2) * RANK + j] = acc[v];
    }
}

// ============================================================
// WMMA over one staged K_CHUNK; wave wv owns 16-col strip,
// all 8 row-tiles.  A/B fragments are single v2f LDS reads.
// ============================================================
__device__ __forceinline__ void mma_chunk(const float* xsb, const float* bsb,
                                          v8f acc[8], int lane, int wv) {
    const int ln = lane & 15;
    const int h2 = (lane >> 4) << 1;
    #pragma unroll
    for (int ks = 0; ks < K_CHUNK; ks += 4) {
        const int kk = ks + h2;                          // always even
        v2f b = *(const v2f*)(bsb + (kk >> 1) * BN_S + (wv * 16 + ln) * 2);
        #pragma unroll
        for (int rt = 0; rt < 8; ++rt) {
            v2f a = *(const v2f*)(xsb + (rt * 16 + ln) * XS_S + kk);
            acc[rt] = __builtin_amdgcn_wmma_f32_16x16x4_f32(false, a, false, b,
                                                            (short)0, acc[rt],
                                                            false, false);
        }
    }
}

// ============================================================
// 6) fused GEMM: out[tok] = x[tok] @ (W_base^T + DeltaW[a]) + r @ B[a]
//    register double-buffered pipeline, one barrier per chunk:
//    store(i) -> barrier -> global-load(i+1) -> wmma(i)
// ============================================================
__global__ __launch_bounds__(256) void k_gemm(const float* __restrict__ x,
                                              const float* __restrict__ Wb,
                                              const float* __restrict__ Dw,
                                              const float* __restrict__ Bbuf,
                                              char* ws,
                                              float* __restrict__ out) {
    __shared__ float xs[2][M_TILE * XS_S];
    __shared__ float bs[2][(K_CHUNK / 2) * BN_S];
    __shared__ int   ts[M_TILE];

    TileDesc td = ((const TileDesc*)(ws + WS_TILES))[blockIdx.y];
    if (td.row_cnt == 0) return;
    const int* sorted = (const int*)(ws + WS_SORTED);
    const float* r    = (const float*)(ws + WS_R);

    const int nb   = blockIdx.x * N_TILE;
    const int tid  = threadIdx.x;
    const int lane = tid & 31;
    const int wv   = tid >> 5;

    // clamped token ids: always-valid gather addresses, no masking needed
    if (tid < M_TILE) ts[tid] = sorted[td.row_start + min(tid, td.row_cnt - 1)];
    __syncthreads();

    v8f acc[8];
    #pragma unroll
    for (int i = 0; i < 8; ++i) acc[i] = (v8f){0.f,0.f,0.f,0.f,0.f,0.f,0.f,0.f};

    // ---- per-thread staging geometry (invariant over chunks)
    const int q    = (tid & 3) << 2;          // 0,4,8,12
    const int rowA = tid >> 2;                // rows rowA and rowA+64
    const long xb0 = (long)ts[rowA]      * DIM + q;
    const long xb1 = (long)ts[rowA + 64] * DIM + q;
    const int  kk_s = tid >> 5;               // 0..7 (k rows kk_s, kk_s+8)
    const int  n4   = (tid & 31) << 2;        // 0..124
    const long adD  = (long)td.adapter * DIM;
    const float* dw0 = Dw + (adD + kk_s) * DIM + nb + n4;
    const float* dw1 = dw0 + 8l * DIM;
    const float* wbB = Wb + (long)(nb + n4) * DIM + kk_s;  // transposed access
    const float* bb0 = Bbuf + ((long)td.adapter * RANK + kk_s) * DIM + nb + n4;
    const float* bb1 = bb0 + 8l * DIM;
    const float* rp0 = r + (long)(td.row_start + min(rowA,      td.row_cnt - 1)) * RANK + q;
    const float* rp1 = r + (long)(td.row_start + min(rowA + 64, td.row_cnt - 1)) * RANK + q;

    const int NC = DIM / K_CHUNK;             // 256 main chunks + 1 rank chunk

    // ---- prefetch chunk 0 into registers
    float4 px0 = *(const float4*)(x + xb0);
    float4 px1 = *(const float4*)(x + xb1);
    float4 pb0 = *(const float4*)(dw0);
    float4 pb1 = *(const float4*)(dw1);
    float4 pwA = make_float4(wbB[0], wbB[DIM], wbB[2 * DIM], wbB[3 * DIM]);
    float4 pwB = make_float4(wbB[8], wbB[8 + DIM], wbB[8 + 2 * DIM], wbB[8 + 3 * DIM]);

    for (int i = 0; i <= NC; ++i) {
        float* xsb = xs[i & 1];
        float* bsb = bs[i & 1];
        // ---- store staged registers to LDS (B_eff = DeltaW + W_base^T)
        *(float4*)(xsb + rowA * XS_S + q)        = px0;
        *(float4*)(xsb + (rowA + 64) * XS_S + q) = px1;
        {
            float* d0 = bsb + (kk_s >> 1) * BN_S + (kk_s & 1);        // kr = kk_s
            d0[(n4 + 0) * 2] = pb0.x + pwA.x;
            d0[(n4 + 1) * 2] = pb0.y + pwA.y;
            d0[(n4 + 2) * 2] = pb0.z + pwA.z;
            d0[(n4 + 3) * 2] = pb0.w + pwA.w;
            int kr1 = kk_s + 8;
            float* d1 = bsb + (kr1 >> 1) * BN_S + (kr1 & 1);
            d1[(n4 + 0) * 2] = pb1.x + pwB.x;
            d1[(n4 + 1) * 2] = pb1.y + pwB.y;
            d1[(n4 + 2) * 2] = pb1.z + pwB.z;
            d1[(n4 + 3) * 2] = pb1.w + pwB.w;
        }
        __syncthreads();
        // ---- issue next chunk's global loads (overlap with WMMA below)
        if (i + 1 < NC) {
            const int k = (i + 1) * K_CHUNK;
            px0 = *(const float4*)(x + xb0 + k);
            px1 = *(const float4*)(x + xb1 + k);
            pb0 = *(const float4*)(dw0 + (long)k * DIM);
            pb1 = *(const float4*)(dw1 + (long)k * DIM);
            const float* w0 = wbB + k;
            pwA = make_float4(w0[0], w0[DIM], w0[2 * DIM], w0[3 * DIM]);
            pwB = make_float4(w0[8], w0[8 + DIM], w0[8 + 2 * DIM], w0[8 + 3 * DIM]);
        } else if (i + 1 == NC) {
            // low-rank chunk: A-frag <- r rows, B-frag <- B_buffer[a]
            px0 = *(const float4*)(rp0);
            px1 = *(const float4*)(rp1);
            pb0 = *(const float4*)(bb0);
            pb1 = *(const float4*)(bb1);
            pwA = make_float4(0.f, 0.f, 0.f, 0.f);
            pwB = pwA;
        }
        mma_chunk(xsb, bsb, acc, lane, wv);
        // no trailing barrier needed: next store targets the other buffer,
        // and the barrier at the top of iteration i+1 orders the reuse.
    }

    // ---- epilogue: scatter-store (each token row owned by exactly one tile)
    const int n  = nb + wv * 16 + (lane & 15);
    const int mh = (lane >> 4) << 3;
    #pragma unroll
    for (int rt = 0; rt < 8; ++rt) {
        #pragma unroll
        for (int v = 0; v < 8; ++v) {
            int m = rt * 16 + v + mh;
            if (m < td.row_cnt)
                out[(long)ts[m] * DIM + n] = acc[rt][v];
        }
    }
}

// ============================================================
extern "C" void kernel_launch(void* const* d_in, const int* in_sizes, int n_in,
                              void* d_out, int out_size, void* d_ws, size_t ws_size,
                              hipStream_t stream) {
    const float* x   = (const float*)d_in[0];
    const int*   idx = (const int*)  d_in[1];
    const float* Wb  = (const float*)d_in[2];
    const float* Ab  = (const float*)d_in[3];
    const float* Bb  = (const float*)d_in[4];
    const float* Dw  = (const float*)d_in[5];
    float* out = (float*)d_out;
    char*  ws  = (char*)d_ws;

    k_zero   <<<1, 32, 0, stream>>>(ws);
    k_count  <<<T_TOK / 256, 256, 0, stream>>>(idx, ws);
    k_scan   <<<1, 1, 0, stream>>>(ws);
    k_scatter<<<T_TOK / 256, 256, 0, stream>>>(idx, ws);
    k_rank   <<<MAX_TILES, 256, 0, stream>>>(x, Ab, ws);
    k_gemm   <<<dim3(DIM / N_TILE, MAX_TILES), 256, 0, stream>>>(x, Wb, Dw, Bb, ws, out);
}